// HybridQNN_88270167867622
// MI455X (gfx1250) — compile-verified
//
#include <hip/hip_runtime.h>

// ---------------------------------------------------------------------------
// Math: full circuit matrix M = L2*L1*L0 (real 4x4), each layer
//   L = P * (I (x) RY(w1)) * (RY(w0) (x) I),  P = combined CNOT01*CNOT10 perm
// Final answer per element reduces to:
//   z1 = K1*cos(x0) + K2*cos(x1) + K3*cos(x0)cos(x1) + K4*sin(x0)sin(x1)
// with K's built from rows 1,3 of M (derivation uses |state|=1 and
// half-angle identities). Memory-bound: 12 B/element, ~50 MB total.
// ---------------------------------------------------------------------------

typedef __attribute__((ext_vector_type(2))) float v2f;
typedef __attribute__((ext_vector_type(8))) float v8f;

// L_i[r][c] given half-angle trig of the layer's two RY angles.
__device__ __forceinline__ float L_entry(int r, int c,
                                         float c0, float s0, float c1, float s1) {
  const float cc = c0 * c1, cs = c0 * s1, sc = s0 * c1, ss = s0 * s1;
  const float m[4][4] = {
      { cc, -cs, -sc,  ss},
      { sc, -ss,  cc, -cs},
      { ss,  sc,  cs,  cc},
      { cs,  cc, -ss, -sc}};
  return m[r & 3][c & 3];
}

// Run-once prologue (1 wave): compose M = L2*L1*L0 with two f32 WMMAs,
// then fold into the 4 streaming coefficients K1..K4 -> coef[0..3].
__global__ void qnn_coef_kernel(const float* __restrict__ w,
                                float* __restrict__ coef) {
  const int lane  = threadIdx.x & 31;
  const int mrow  = lane & 15;          // A row index / B column index
  const int khalf = (lane >> 4) << 1;   // K pair base: 0 (lanes 0-15) or 2

  float hc[3], hs[3], gc[3], gs[3];
#pragma unroll
  for (int i = 0; i < 3; ++i) {
    hc[i] = cosf(0.5f * w[2 * i]);     hs[i] = sinf(0.5f * w[2 * i]);
    gc[i] = cosf(0.5f * w[2 * i + 1]); gs[i] = sinf(0.5f * w[2 * i + 1]);
  }

  const bool ok = (mrow < 4);
  // WMMA #1: D1 = L1 * L0   (A: 16x4 f32 layout, B: 4x16, zero-padded)
  v2f A, B;
  A.x = ok ? L_entry(mrow, khalf + 0, hc[1], hs[1], gc[1], gs[1]) : 0.f;
  A.y = ok ? L_entry(mrow, khalf + 1, hc[1], hs[1], gc[1], gs[1]) : 0.f;
  B.x = ok ? L_entry(khalf + 0, mrow, hc[0], hs[0], gc[0], gs[0]) : 0.f;
  B.y = ok ? L_entry(khalf + 1, mrow, hc[0], hs[0], gc[0], gs[0]) : 0.f;
  v8f Z = {};
  v8f D1 = __builtin_amdgcn_wmma_f32_16x16x4_f32(false, A, false, B,
                                                 (short)0, Z, false, false);

  // Re-layout D1 (C/D layout) into a B operand: B2[k][n] = D1[k][n].
  // lanes 0-15 already hold rows 0,1 at the right lanes; rows 2,3 need a
  // cross-half shuffle into lanes 16-31.
  float b2x_hi = __shfl(D1[2], mrow);
  float b2y_hi = __shfl(D1[3], mrow);
  v2f B2, A2;
  B2.x = (lane < 16) ? D1[0] : b2x_hi;
  B2.y = (lane < 16) ? D1[1] : b2y_hi;
  A2.x = ok ? L_entry(mrow, khalf + 0, hc[2], hs[2], gc[2], gs[2]) : 0.f;
  A2.y = ok ? L_entry(mrow, khalf + 1, hc[2], hs[2], gc[2], gs[2]) : 0.f;
  // WMMA #2: M = L2 * (L1*L0)
  v8f M = __builtin_amdgcn_wmma_f32_16x16x4_f32(false, A2, false, B2,
                                                (short)0, Z, false, false);

  // Rows 1 and 3 of M, columns 0..3 (D layout: VGPR r, lanes 0-15 = row r).
  const float M10 = __shfl(M[1], 0), M11 = __shfl(M[1], 1);
  const float M12 = __shfl(M[1], 2), M13 = __shfl(M[1], 3);
  const float M30 = __shfl(M[3], 0), M31 = __shfl(M[3], 1);
  const float M32 = __shfl(M[3], 2), M33 = __shfl(M[3], 3);

  const float g00 = M10 * M10 + M30 * M30;   // |Re|^2 coeffs on (a,d)
  const float g11 = M13 * M13 + M33 * M33;
  const float h00 = M11 * M11 + M31 * M31;   // |Im|^2 coeffs on (b,c)
  const float h11 = M12 * M12 + M32 * M32;
  const float K4  = (M10 * M13 + M30 * M33) - (M11 * M12 + M31 * M32);
  const float K1  = -0.5f * (g00 - g11 + h00 - h11);
  const float K2  = -0.5f * (g00 - g11 - h00 + h11);
  const float K3  = -0.5f * (g00 + g11 - h00 - h11);

  if (lane == 0) {
    coef[0] = K1; coef[1] = K2; coef[2] = K3; coef[3] = K4;
  }
}

__device__ __forceinline__ float qeval(float x0, float x1,
                                       float K1, float K2, float K3, float K4) {
  float s0, c0, s1, c1;
  __sincosf(x0, &s0, &c0);   // v_sin_f32 / v_cos_f32 (args ~N(0,1), in range)
  __sincosf(x1, &s1, &c1);
  return fmaf(K1, c0, fmaf(K2, c1, fmaf(K3, c0 * c1, K4 * (s0 * s1))));
}

// Streaming main kernel: 4 elements/thread, 2x b128 loads + 1x b128 store.
// Default (RT) caching on purpose: 50 MB working set fits the 192 MB L2.
__global__ __launch_bounds__(256) void qnn_main_kernel(
    const float* __restrict__ x, const float* __restrict__ coef,
    float* __restrict__ out, int n) {
  const float K1 = coef[0], K2 = coef[1], K3 = coef[2], K4 = coef[3];
  const int t = blockIdx.x * blockDim.x + threadIdx.x;
  const int base = t * 4;  // first element index for this thread
  if (base + 3 < n) {
    const float4 xa = ((const float4*)x)[t * 2 + 0];  // elems base, base+1
    const float4 xb = ((const float4*)x)[t * 2 + 1];  // elems base+2, base+3
    float4 r;
    r.x = qeval(xa.x, xa.y, K1, K2, K3, K4);
    r.y = qeval(xa.z, xa.w, K1, K2, K3, K4);
    r.z = qeval(xb.x, xb.y, K1, K2, K3, K4);
    r.w = qeval(xb.z, xb.w, K1, K2, K3, K4);
    ((float4*)out)[t] = r;
  } else {
    for (int e = base; e < n; ++e)
      out[e] = qeval(x[2 * e], x[2 * e + 1], K1, K2, K3, K4);
  }
}

extern "C" void kernel_launch(void* const* d_in, const int* in_sizes, int n_in,
                              void* d_out, int out_size, void* d_ws, size_t ws_size,
                              hipStream_t stream) {
  const float* x = (const float*)d_in[0];   // (B, 2) float32
  const float* w = (const float*)d_in[1];   // (6,)  float32
  float* out  = (float*)d_out;              // (B,)  float32
  float* coef = (float*)d_ws;               // 4 floats of scratch

  const int B = in_sizes[0] / 2;

  qnn_coef_kernel<<<1, 32, 0, stream>>>(w, coef);

  const int nThreads = (B + 3) / 4;
  const int block = 256;
  const int grid = (nThreads + block - 1) / block;
  qnn_main_kernel<<<grid, block, 0, stream>>>(x, coef, out, B);
}